// StockGraphModel_1116691497177
// MI455X (gfx1250) — compile-verified
//
#include <hip/hip_runtime.h>

#define N_NODES 100000
#define N_EDGES 1600000
#define IN_C    128
#define HID     256
#define OUT_C   32
#define EPSF    1e-5f
#define SLOPE   0.1f

typedef float v2f __attribute__((ext_vector_type(2)));
typedef float v8f __attribute__((ext_vector_type(8)));

// ---------------------------------------------------------------- utilities
__global__ void fill_f32(float* __restrict__ p, float v, int n) {
    int i = blockIdx.x * blockDim.x + threadIdx.x;
    if (i < n) p[i] = v;
}

__global__ void edge_deg(const int* __restrict__ dst, const float* __restrict__ ew,
                         float* __restrict__ deg, int e) {
    int i = blockIdx.x * blockDim.x + threadIdx.x;
    if (i < e) atomicAdd(&deg[dst[i]], ew[i]);
}

__global__ void deg_to_dinv(float* __restrict__ d, int n) {
    int i = blockIdx.x * blockDim.x + threadIdx.x;
    if (i < n) {
        float v = d[i];
        d[i] = (v > 0.0f) ? rsqrtf(fmaxf(v, 1e-12f)) : 0.0f;
    }
}

__global__ void edge_norm(const int* __restrict__ src, const int* __restrict__ dst,
                          const float* __restrict__ ew, const float* __restrict__ dinv,
                          float* __restrict__ norm, int e) {
    int i = blockIdx.x * blockDim.x + threadIdx.x;
    if (i < e) norm[i] = dinv[src[i]] * ew[i] * dinv[dst[i]];
}

// ------------------------------------------------- f32 WMMA GEMM: Y = X @ W (+b, leaky)
// One wave computes a 16 x (16*CT) strip of Y. A operand (16x4 f32) is reused
// across the CT column tiles; K stepped by 4 per v_wmma_f32_16x16x4_f32.
template<int FIN, int FOUT, int CT, bool BIAS, bool LEAKY>
__global__ __launch_bounds__(256) void gemm_wmma_f32(
    const float* __restrict__ X, const float* __restrict__ W,
    const float* __restrict__ bias, float* __restrict__ Y, int nrows)
{
    const int waveId = (blockIdx.x * blockDim.x + threadIdx.x) >> 5;
    const int lane   = threadIdx.x & 31;
    const int wTiles = FOUT / (16 * CT);          // wave col-groups per row block
    const int rowTile = waveId / wTiles;
    const int colGrp  = waveId % wTiles;
    const int row0 = rowTile * 16;
    if (row0 >= nrows) return;
    const int col0 = colGrp * 16 * CT;
    const int half = lane >> 4;                   // K-pair selector
    const int lid  = lane & 15;                   // M (for A) / N (for B,C)

    v8f acc[CT];
#pragma unroll
    for (int c = 0; c < CT; ++c)
#pragma unroll
        for (int j = 0; j < 8; ++j) acc[c][j] = 0.0f;

    // A layout (16x4 f32): lanes 0-15 rows, v0/v1 = K {0,1} (half 0) or {2,3} (half 1)
    const float* __restrict__ arow = X + (size_t)(row0 + lid) * FIN + 2 * half;
    // B layout (4x16 f32): lane = N, v0/v1 = K {0,1}/{2,3}
    const float* __restrict__ wbase = W + col0 + lid;

#pragma unroll 2
    for (int k0 = 0; k0 < FIN; k0 += 4) {
        v2f a;
        a.x = arow[k0];
        a.y = arow[k0 + 1];
        const float* wk = wbase + (size_t)(k0 + 2 * half) * FOUT;
#pragma unroll
        for (int c = 0; c < CT; ++c) {
            v2f b;
            b.x = wk[c * 16];
            b.y = wk[c * 16 + FOUT];
            acc[c] = __builtin_amdgcn_wmma_f32_16x16x4_f32(
                false, a, false, b, (short)0, acc[c], false, false);
        }
    }

    // C/D layout: VGPR r, lanes 0-15 -> M=r, lanes 16-31 -> M=8+r
#pragma unroll
    for (int c = 0; c < CT; ++c) {
        const int col = col0 + c * 16 + lid;
        const float bv = BIAS ? bias[col] : 0.0f;
        float* yp = Y + (size_t)(row0 + 8 * half) * FOUT + col;
#pragma unroll
        for (int r = 0; r < 8; ++r) {
            float v = acc[c][r] + bv;
            if (LEAKY) v = (v >= 0.0f) ? v : SLOPE * v;
            yp[(size_t)r * FOUT] = v;
        }
    }
}

// ------------------------------------------------------------ GCN aggregation
// AGG = dinv^2 * H + bias   (self-loop term + bias folded into the init pass)
template<int F>
__global__ void init_agg(const float* __restrict__ h, const float* __restrict__ dinv,
                         const float* __restrict__ bias, float* __restrict__ agg, int n) {
    size_t i = (size_t)blockIdx.x * blockDim.x + threadIdx.x;
    if (i >= (size_t)n * F) return;
    int node = (int)(i / F);
    int f    = (int)(i % F);
    float di = dinv[node];
    agg[i] = di * di * h[i] + bias[f];
}

// AGG[dst] += norm[e] * H[src]; one block per edge, coalesced 1KB row, L2 atomics
template<int F>
__global__ __launch_bounds__(F) void aggregate(
    const int* __restrict__ src, const int* __restrict__ dst,
    const float* __restrict__ norm, const float* __restrict__ h,
    float* __restrict__ agg)
{
    const int e = blockIdx.x;
    const int f = threadIdx.x;
    const int s = src[e];       // uniform -> scalar load
    const int d = dst[e];
    const float w = norm[e];
    if (e + 1 < (int)gridDim.x) // hide gather latency for next edge's row
        __builtin_prefetch(h + (size_t)src[e + 1] * F + f, 0, 0);
    atomicAdd(&agg[(size_t)d * F + f], h[(size_t)s * F + f] * w);
}

// ------------------------------------------------------------ BatchNorm
template<int F>
__global__ __launch_bounds__(F) void col_stats(const float* __restrict__ x,
                                               float* __restrict__ sums,
                                               int n, int rowsPerBlk) {
    const int f  = threadIdx.x;
    const int r0 = blockIdx.x * rowsPerBlk;
    const int r1 = min(r0 + rowsPerBlk, n);
    float s = 0.0f, q = 0.0f;
    for (int r = r0; r < r1; ++r) {
        float v = x[(size_t)r * F + f];   // fully coalesced row reads
        s += v; q += v * v;
    }
    atomicAdd(&sums[f], s);
    atomicAdd(&sums[F + f], q);
}

template<int F>
__global__ void bn_params(const float* __restrict__ sums, const float* __restrict__ g,
                          const float* __restrict__ be, float* __restrict__ sc, int n) {
    int f = threadIdx.x;
    if (f >= F) return;
    float inv_n = 1.0f / (float)n;
    float mean  = sums[f] * inv_n;
    float var   = sums[F + f] * inv_n - mean * mean;
    float scale = g[f] * rsqrtf(var + EPSF);
    sc[f]     = scale;
    sc[F + f] = be[f] - mean * scale;
}

template<int F>
__global__ void bn_apply_leaky(const float* __restrict__ x, const float* __restrict__ sc,
                               float* __restrict__ y, int n) {
    size_t i = (size_t)blockIdx.x * blockDim.x + threadIdx.x;
    if (i >= (size_t)n * F) return;
    int f = (int)(i % F);
    float v = x[i] * sc[f] + sc[F + f];
    y[i] = (v >= 0.0f) ? v : SLOPE * v;
}

// ---------------------------------------------------------------- host side
template<int FIN>
static void run_conv(const float* xin, const float* Wm, const float* b,
                     const float* g, const float* be,
                     const int* src, const int* dst,
                     const float* dinv, const float* norm,
                     float* H, float* AGG, float* ACT,
                     float* stats, float* bnsc, hipStream_t stream)
{
    // H = xin @ W  (no bias yet)
    {
        const int waves  = (N_NODES / 16) * (HID / 64);
        const int blocks = (waves * 32 + 255) / 256;
        gemm_wmma_f32<FIN, HID, 4, false, false>
            <<<blocks, 256, 0, stream>>>(xin, Wm, nullptr, H, N_NODES);
    }
    // AGG = dinv^2 * H + bias
    {
        size_t tot = (size_t)N_NODES * HID;
        init_agg<HID><<<(int)((tot + 255) / 256), 256, 0, stream>>>(H, dinv, b, AGG, N_NODES);
    }
    // AGG[dst] += norm * H[src]
    aggregate<HID><<<N_EDGES, HID, 0, stream>>>(src, dst, norm, H, AGG);
    // BatchNorm (training-mode stats) + LeakyReLU -> ACT
    fill_f32<<<(2 * HID + 255) / 256, 256, 0, stream>>>(stats, 0.0f, 2 * HID);
    {
        const int nblk = 512;
        const int rpb  = (N_NODES + nblk - 1) / nblk;
        col_stats<HID><<<nblk, HID, 0, stream>>>(AGG, stats, N_NODES, rpb);
    }
    bn_params<HID><<<1, HID, 0, stream>>>(stats, g, be, bnsc, N_NODES);
    {
        size_t tot = (size_t)N_NODES * HID;
        bn_apply_leaky<HID><<<(int)((tot + 255) / 256), 256, 0, stream>>>(AGG, bnsc, ACT, N_NODES);
    }
}

extern "C" void kernel_launch(void* const* d_in, const int* in_sizes, int n_in,
                              void* d_out, int out_size, void* d_ws, size_t ws_size,
                              hipStream_t stream) {
    (void)in_sizes; (void)n_in; (void)out_size; (void)ws_size;

    const float* x    = (const float*)d_in[0];
    const int*   ei   = (const int*)  d_in[1];
    const float* ew   = (const float*)d_in[2];
    const float* W1   = (const float*)d_in[3];
    const float* b1   = (const float*)d_in[4];
    const float* g1   = (const float*)d_in[5];
    const float* be1  = (const float*)d_in[6];
    const float* W2   = (const float*)d_in[7];
    const float* b2   = (const float*)d_in[8];
    const float* g2   = (const float*)d_in[9];
    const float* be2  = (const float*)d_in[10];
    const float* W3   = (const float*)d_in[11];
    const float* b3   = (const float*)d_in[12];
    const float* g3   = (const float*)d_in[13];
    const float* be3  = (const float*)d_in[14];
    const float* Wl1  = (const float*)d_in[15];
    const float* bl1  = (const float*)d_in[16];
    const float* Wl2  = (const float*)d_in[17];
    const float* bl2  = (const float*)d_in[18];

    const int* src = ei;              // edge_index[0, :]
    const int* dst = ei + N_EDGES;    // edge_index[1, :]

    // carve workspace
    char* ws = (char*)d_ws;
    auto carve = [&](size_t bytes) -> float* {
        float* p = (float*)ws;
        ws += (bytes + 255) & ~(size_t)255;
        return p;
    };
    const size_t big = (size_t)N_NODES * HID * sizeof(float);   // ~102.4 MB
    float* H     = carve(big);
    float* AGG   = carve(big);
    float* ACT   = carve(big);
    float* dinv  = carve((size_t)N_NODES * sizeof(float));
    float* norm  = carve((size_t)N_EDGES * sizeof(float));
    float* stats = carve(2 * HID * sizeof(float));
    float* bnsc  = carve(2 * HID * sizeof(float));

    // degree -> D^-1/2 -> per-edge norm
    fill_f32<<<(N_NODES + 255) / 256, 256, 0, stream>>>(dinv, 1.0f, N_NODES); // self loops
    edge_deg<<<(N_EDGES + 255) / 256, 256, 0, stream>>>(dst, ew, dinv, N_EDGES);
    deg_to_dinv<<<(N_NODES + 255) / 256, 256, 0, stream>>>(dinv, N_NODES);
    edge_norm<<<(N_EDGES + 255) / 256, 256, 0, stream>>>(src, dst, ew, dinv, norm, N_EDGES);

    // three GCN conv + BN + leaky layers
    run_conv<IN_C>(x,   W1, b1, g1, be1, src, dst, dinv, norm, H, AGG, ACT, stats, bnsc, stream);
    run_conv<HID >(ACT, W2, b2, g2, be2, src, dst, dinv, norm, H, AGG, ACT, stats, bnsc, stream);
    run_conv<HID >(ACT, W3, b3, g3, be3, src, dst, dinv, norm, H, AGG, ACT, stats, bnsc, stream);

    // lin1: H = leaky(ACT @ Wl1 + bl1)   (bias + leaky fused into WMMA epilogue)
    {
        const int waves  = (N_NODES / 16) * (HID / 64);
        const int blocks = (waves * 32 + 255) / 256;
        gemm_wmma_f32<HID, HID, 4, true, true>
            <<<blocks, 256, 0, stream>>>(ACT, Wl1, bl1, H, N_NODES);
    }
    // lin2: out = H @ Wl2 + bl2   (FOUT=32 -> CT=2, one col-group per row tile)
    {
        const int waves  = (N_NODES / 16) * (OUT_C / 32);
        const int blocks = (waves * 32 + 255) / 256;
        gemm_wmma_f32<HID, OUT_C, 2, true, false>
            <<<blocks, 256, 0, stream>>>(H, Wl2, bl2, (float*)d_out, N_NODES);
    }
}